// CustomLSTM_76063870812452
// MI455X (gfx1250) — compile-verified
//
#include <hip/hip_runtime.h>

typedef __attribute__((ext_vector_type(2))) float v2f;
typedef __attribute__((ext_vector_type(8))) float v8f;

// V_WMMA_F32_16X16X4_F32: D = A(16x4,f32) x B(4x16,f32) + C(16x16,f32)
#define WMMA_F32(a, b, c) \
  __builtin_amdgcn_wmma_f32_16x16x4_f32(false, (a), false, (b), (short)0, (c), false, false)

static constexpr int kB  = 64;
static constexpr int kT  = 512;
static constexpr int kH  = 1024;
static constexpr int kG  = 4096;          // 4H
static constexpr int kBT = kB * kT;       // 32768 rows for the input projections

// ---------------------------------------------------------------------------
// Zero h ping-pong buffers and grid-barrier counters (runs once per call).
// ---------------------------------------------------------------------------
__global__ void init_kernel(float* hbufs, unsigned int* cnts, int n) {
  const int i = blockIdx.x * blockDim.x + threadIdx.x;
  if (i < n) hbufs[i] = 0.0f;
  if (i < 2) cnts[i] = 0u;
}

// ---------------------------------------------------------------------------
// out[M,N] = A[M,K] * W[N,K]^T + bias1[N] + bias2[N]
// One wave computes a 64x64 tile: 4x4 grid of 16x16 WMMA accumulators.
// ---------------------------------------------------------------------------
__global__ __launch_bounds__(128) void gemm_bias_kernel(
    const float* __restrict__ A, const float* __restrict__ W,
    const float* __restrict__ bias1, const float* __restrict__ bias2,
    float* __restrict__ out, int M, int N, int K) {
  const int lane = threadIdx.x & 31;
  const int sub  = lane >> 4;   // lane half selects K pair (A) / K rows (B)
  const int l16  = lane & 15;   // M index (A) / N index (B, C/D)
  const int wid  = blockIdx.x * (blockDim.x >> 5) + (threadIdx.x >> 5);
  const int tilesN = N >> 6;
  const int tm = wid / tilesN;
  const int tn = wid - tm * tilesN;
  const int m0 = tm << 6;
  const int n0 = tn << 6;

  const float* Ar[4];
  const float* Wr[4];
#pragma unroll
  for (int i = 0; i < 4; ++i) {
    Ar[i] = A + (size_t)(m0 + i * 16 + l16) * K + 2 * sub;
    Wr[i] = W + (size_t)(n0 + i * 16 + l16) * K + 2 * sub;
  }

  v8f acc[4][4] = {};
  for (int k = 0; k < K; k += 4) {
    v2f a[4], b[4];
#pragma unroll
    for (int i = 0; i < 4; ++i) a[i] = *(const v2f*)(Ar[i] + k);
#pragma unroll
    for (int j = 0; j < 4; ++j) b[j] = *(const v2f*)(Wr[j] + k);
#pragma unroll
    for (int i = 0; i < 4; ++i)
#pragma unroll
      for (int j = 0; j < 4; ++j)
        acc[i][j] = WMMA_F32(a[i], b[j], acc[i][j]);
  }

#pragma unroll
  for (int j = 0; j < 4; ++j) {
    const int n = n0 + j * 16 + l16;
    const float bs = bias1[n] + bias2[n];
#pragma unroll
    for (int i = 0; i < 4; ++i) {
      const int mb = m0 + i * 16 + 8 * sub;   // C/D layout: M = v + 8*(lane>>4)
#pragma unroll
      for (int v = 0; v < 8; ++v)
        out[(size_t)(mb + v) * N + n] = acc[i][j][v] + bs;
    }
  }
}

// ---------------------------------------------------------------------------
// Persistent LSTM recurrence for one layer.
// 64 workgroups x 128 threads. wg g owns hidden slice [16g, 16g+16);
// wave w in {0..3} owns gate block w (i,f,g,o). One grid barrier per step.
// ---------------------------------------------------------------------------
__global__ __launch_bounds__(128) void lstm_rec_kernel(
    const float* __restrict__ XG,   // [B*T, 4H], row = b*T + t (bias baked in)
    const float* __restrict__ Whh,  // [4H, H] row-major
    float* hping,                   // two [B,H] ping-pong buffers (pre-zeroed)
    float* __restrict__ out,        // [B, T, H]
    float* __restrict__ hn,         // [B, H] final hidden
    float* __restrict__ cn,         // [B, H] final cell
    unsigned int* cnt, int T) {
  __shared__ float gact[4][64][16];   // post-activation gates
  __shared__ float cbuf[64][16];      // persistent cell state for this slice

  const int g    = blockIdx.x;
  const int w    = threadIdx.x >> 5;      // gate block (wave-uniform)
  const int lane = threadIdx.x & 31;
  const int sub  = lane >> 4;
  const int l16  = lane & 15;
  const int j0   = g * 16;
  const int nglb = w * kH + j0 + l16;     // W_hh row / gate column

  for (int e = threadIdx.x; e < 64 * 16; e += 128) cbuf[e >> 4][e & 15] = 0.0f;
  __syncthreads();

  const float*   Wr  = Whh + (size_t)nglb * kH + 2 * sub;
  const unsigned nwg = gridDim.x;

  for (int t = 0; t < T; ++t) {
    const float* hr = hping + (size_t)(t & 1) * (kB * kH);
    float*       hw = hping + (size_t)((t + 1) & 1) * (kB * kH);

    // gates(64 x 16) = h(64 x 1024) * Whh_slice^T, as 4 stacked 16x16 tiles
    v8f acc[4] = {};
    const float* hA = hr + (size_t)l16 * kH + 2 * sub;
    for (int k = 0; k < kH; k += 4) {
      const v2f b  = *(const v2f*)(Wr + k);
      const v2f a0 = *(const v2f*)(hA + k);
      const v2f a1 = *(const v2f*)(hA + (size_t)16 * kH + k);
      const v2f a2 = *(const v2f*)(hA + (size_t)32 * kH + k);
      const v2f a3 = *(const v2f*)(hA + (size_t)48 * kH + k);
      acc[0] = WMMA_F32(a0, b, acc[0]);
      acc[1] = WMMA_F32(a1, b, acc[1]);
      acc[2] = WMMA_F32(a2, b, acc[2]);
      acc[3] = WMMA_F32(a3, b, acc[3]);
    }

    // add precomputed x-projection, activate, exchange through LDS
#pragma unroll
    for (int mc = 0; mc < 4; ++mc) {
#pragma unroll
      for (int v = 0; v < 8; ++v) {
        const int m = mc * 16 + v + 8 * sub;                    // batch index
        const float x = acc[mc][v] + XG[((size_t)m * T + t) * kG + nglb];
        const float r = (w == 2) ? tanhf(x) : (1.0f / (1.0f + __expf(-x)));
        gact[w][m][l16] = r;
      }
    }
    __syncthreads();

    // c/h update for this wg's 64x16 slice (8 elements per thread)
    for (int e = threadIdx.x; e < 64 * 16; e += 128) {
      const int b_ = e >> 4, jl = e & 15;
      const float iv = gact[0][b_][jl];
      const float fv = gact[1][b_][jl];
      const float gv = gact[2][b_][jl];
      const float ov = gact[3][b_][jl];
      const float c  = fv * cbuf[b_][jl] + iv * gv;
      cbuf[b_][jl]   = c;
      const float hv = ov * tanhf(c);
      const int j    = j0 + jl;
      out[((size_t)b_ * T + t) * kH + j] = hv;
      hw[(size_t)b_ * kH + j]            = hv;
      if (t == T - 1) {
        hn[(size_t)b_ * kH + j] = hv;
        cn[(size_t)b_ * kH + j] = c;
      }
    }

    // device-wide barrier: monotonic counter, one arrive per wg per step
    __threadfence();              // release h_t writes
    __syncthreads();
    if (threadIdx.x == 0) {
      __hip_atomic_fetch_add(cnt, 1u, __ATOMIC_RELEASE, __HIP_MEMORY_SCOPE_AGENT);
      const unsigned tgt = nwg * (unsigned)(t + 1);
      while (__hip_atomic_load(cnt, __ATOMIC_ACQUIRE, __HIP_MEMORY_SCOPE_AGENT) < tgt)
        __builtin_amdgcn_s_sleep(2);
    }
    __syncthreads();
    __threadfence();              // acquire: invalidate stale h in near caches
  }
}

// ---------------------------------------------------------------------------
extern "C" void kernel_launch(void* const* d_in, const int* in_sizes, int n_in,
                              void* d_out, int out_size, void* d_ws, size_t ws_size,
                              hipStream_t stream) {
  (void)in_sizes; (void)n_in; (void)out_size; (void)ws_size;
  const float* X    = (const float*)d_in[0];
  const float* Wih0 = (const float*)d_in[1];
  const float* bih0 = (const float*)d_in[2];
  const float* Whh0 = (const float*)d_in[3];
  const float* bhh0 = (const float*)d_in[4];
  const float* Wih1 = (const float*)d_in[5];
  const float* bih1 = (const float*)d_in[6];
  const float* Whh1 = (const float*)d_in[7];
  const float* bhh1 = (const float*)d_in[8];
  float* outp = (float*)d_out;

  // workspace layout (floats): XG [32768*4096] | out0 [32768*1024] | 4 h bufs | counters
  float* ws   = (float*)d_ws;
  float* XGb  = ws;
  float* out0 = ws + (size_t)kBT * kG;
  float* hb   = out0 + (size_t)kBT * kH;
  unsigned int* cnts = (unsigned int*)(hb + 4 * kB * kH);

  // output layout: out1 [B,T,H] | h_n [2,B,H] | c_n [2,B,H]
  float* out1 = outp;
  float* hn0  = outp + (size_t)kBT * kH;
  float* hn1  = hn0 + kB * kH;
  float* cn0  = hn0 + 2 * kB * kH;
  float* cn1  = cn0 + kB * kH;

  const int nInit = 4 * kB * kH;
  init_kernel<<<(nInit + 255) / 256, 256, 0, stream>>>(hb, cnts, nInit);

  const int gemmBlocks = (kBT / 64) * (kG / 64) / 4;   // 8192 blocks, 4 waves each

  // layer 0
  gemm_bias_kernel<<<gemmBlocks, 128, 0, stream>>>(X, Wih0, bih0, bhh0, XGb, kBT, kG, kH);
  lstm_rec_kernel<<<64, 128, 0, stream>>>(XGb, Whh0, hb, out0, hn0, cn0, cnts + 0, kT);

  // layer 1 (input projection consumes out0; XG buffer reused)
  gemm_bias_kernel<<<gemmBlocks, 128, 0, stream>>>(out0, Wih1, bih1, bhh1, XGb, kBT, kG, kH);
  lstm_rec_kernel<<<64, 128, 0, stream>>>(XGb, Whh1, hb + 2 * kB * kH, out1, hn1, cn1, cnts + 1, kT);
}